// GradientPenaltyLoss_29197187678598
// MI455X (gfx1250) — compile-verified
//
#include <hip/hip_runtime.h>

// Problem constants (from the reference)
#define B_    32
#define N_    68
#define H_    512
#define W_    512
#define RAD   10
#define WIN   (2 * RAD + 1)        // 21
#define GW    (2 * RAD + 3)        // 23  (Sobel halo)
#define GP    24                   // LDS pitch (avoid bank pathologies)
#define NBN   (B_ * N_)            // 2176

typedef float v2f __attribute__((ext_vector_type(2)));
typedef float v8f __attribute__((ext_vector_type(8)));

// Exact-precision 32-lane sum using V_WMMA_F32_16X16X4_F32.
// A-matrix (16x4 f32) layout: lanes 0-15 hold {K=0,K=1}, lanes 16-31 hold {K=2,K=3},
// row M = lane % 16. We place each lane's partial in its first A VGPR (others 0),
// B = all-ones, C = 0  =>  D[m][n] = partial[m] + partial[m+16] for every n.
// C/D layout: c[r] at lane l is row (r + (l<16 ? 0 : 8)). Summing c[0..7] gives
// rows 0-7 on lanes<16 and rows 8-15 on lanes>=16; one xor-16 shuffle finishes.
__device__ __forceinline__ float wave32_sum_wmma(float partial) {
    v2f a;     a.x = partial; a.y = 0.0f;
    v2f ones;  ones.x = 1.0f; ones.y = 1.0f;
    v8f c = {};
    c = __builtin_amdgcn_wmma_f32_16x16x4_f32(
            /*neg_a=*/false, a, /*neg_b=*/false, ones,
            /*c_mod=*/(short)0, c, /*reuse_a=*/false, /*reuse_b=*/false);
    float s = c[0] + c[1] + c[2] + c[3] + c[4] + c[5] + c[6] + c[7];
    s += __shfl_xor(s, 16, 32);
    return s;
}

// One wave (32 threads) per landmark. Stages the 23x23 grayscale patch in LDS
// (zero outside the image = Sobel zero-padding), evaluates |grad| on the 21x21
// window with the boundary mask, reduces via WMMA, writes per-landmark result.
__global__ __launch_bounds__(32)
void GradientPenaltyLoss_landmark_kernel(const float* __restrict__ pred,
                                         const float* __restrict__ images,
                                         const float* __restrict__ vis,
                                         float* __restrict__ ws) {
    __shared__ float lg[GW * GP];

    const int bn  = blockIdx.x;
    const int tid = threadIdx.x;
    const int b   = bn / N_;

    const float lx = pred[bn * 2 + 0];
    const float ly = pred[bn * 2 + 1];
    int px = (int)truncf(lx * (float)W_);
    int py = (int)truncf(ly * (float)H_);
    px = min(max(px, 0), W_ - 1);
    py = min(max(py, 0), H_ - 1);

    const float visible = (vis[bn] >= 0.5f) ? 1.0f : 0.0f;
    if (visible == 0.0f) {               // wave-uniform: local_mean * 0 == 0
        if (tid == 0) ws[bn] = 0.0f;
        return;
    }

    // Stage grayscale patch: rows py-11..py+11, cols px-11..px+11
    const float* img = images + (size_t)b * 3 * H_ * W_;
    const int y0 = py - (RAD + 1);
    const int x0 = px - (RAD + 1);
    for (int i = tid; i < GW * GW; i += 32) {
        const int r = i / GW, cc = i - r * GW;
        const int y = y0 + r, x = x0 + cc;
        float g = 0.0f;
        if ((unsigned)y < (unsigned)H_ && (unsigned)x < (unsigned)W_) {
            const size_t o = (size_t)y * W_ + x;
            g = 0.299f * img[o] + 0.587f * img[o + H_ * W_] + 0.114f * img[o + 2 * H_ * W_];
        }
        lg[r * GP + cc] = g;
    }
    __syncthreads();

    // Sobel magnitude over the 21x21 window (masked to in-image positions).
    float psum = 0.0f;
    for (int i = tid; i < WIN * WIN; i += 32) {
        const int dy = i / WIN, dx = i - dy * WIN;
        const int y = py - RAD + dy, x = px - RAD + dx;
        if ((unsigned)y < (unsigned)H_ && (unsigned)x < (unsigned)W_) {
            const int r = dy + 1, cc = dx + 1;
            const float* rm = &lg[(r - 1) * GP + cc];
            const float* r0 = &lg[(r    ) * GP + cc];
            const float* rp = &lg[(r + 1) * GP + cc];
            const float tl = rm[-1], tc = rm[0], tr = rm[1];
            const float ml = r0[-1],              mr = r0[1];
            const float bl = rp[-1], bc = rp[0], br = rp[1];
            // XLA conv_general_dilated = cross-correlation (no flip)
            const float gx = (tr - tl) + 2.0f * (mr - ml) + (br - bl);
            const float gy = (bl - tl) + 2.0f * (bc - tc) + (br - tr);
            psum += sqrtf(gx * gx + gy * gy);
        }
    }

    const float s = wave32_sum_wmma(psum);

    if (tid == 0) {
        // Boundary-mask count is separable: no reduction needed.
        const int cy = min(py, RAD) + min(H_ - 1 - py, RAD) + 1;
        const int cx = min(px, RAD) + min(W_ - 1 - px, RAD) + 1;
        ws[bn] = (s / (float)(cy * cx)) * visible;
    }
}

// Deterministic final reduction: single wave, fixed accumulation order,
// WMMA cross-lane sum, then the 1/(B*N + 1e-8) scale (PENALTY_WEIGHT = 1).
__global__ __launch_bounds__(32)
void GradientPenaltyLoss_reduce_kernel(const float* __restrict__ ws,
                                       float* __restrict__ out) {
    float p = 0.0f;
    for (int i = threadIdx.x; i < NBN; i += 32) p += ws[i];
    const float s = wave32_sum_wmma(p);
    if (threadIdx.x == 0) out[0] = s * (1.0f / ((float)NBN + 1e-8f));
}

extern "C" void kernel_launch(void* const* d_in, const int* in_sizes, int n_in,
                              void* d_out, int out_size, void* d_ws, size_t ws_size,
                              hipStream_t stream) {
    const float* pred   = (const float*)d_in[0];   // (B, N, 2) f32
    const float* images = (const float*)d_in[1];   // (B, 3, H, W) f32
    const float* vis    = (const float*)d_in[2];   // (B, N) f32
    float* ws  = (float*)d_ws;                     // NBN floats of scratch
    float* out = (float*)d_out;                    // scalar f32

    GradientPenaltyLoss_landmark_kernel<<<NBN, 32, 0, stream>>>(pred, images, vis, ws);
    GradientPenaltyLoss_reduce_kernel<<<1, 32, 0, stream>>>(ws, out);
}